// GenDecAttnBlock_12799002542135
// MI455X (gfx1250) — compile-verified
//
#include <hip/hip_runtime.h>
#include <hip/hip_bf16.h>
#include <stdint.h>

// GenDecAttnBlock on gfx1250 (MI455X).
// Attention block folded into G(256x256) per-pixel GEMM + implicit-GEMM 3x3 conv,
// both on V_WMMA_F32_16X16X4_F32. Conv stages activations in LDS (async-to-LDS
// path when available) shared by all 8 waves, and uses repacked weights for
// contiguous b64 A-fragment loads. conv_b skipped (cancels in instance norm).
//
// Workspace layout (floats):
//   G    : 256*256        =    65536
//   e    : 4*256          =     1024
//   ssum : 4*256          =     1024
//   Arep : 9*128*512      =   589824
//   x1   : 4*256*4096     =  4194304
//   y    : 4*128*4096     =  2097152
//   total ~27.8 MB

typedef __attribute__((ext_vector_type(2))) float v2f;
typedef __attribute__((ext_vector_type(8))) float v8f;

#define WH_F 4096.0f

#if defined(__has_builtin)
#  if __has_builtin(__builtin_amdgcn_global_load_async_to_lds_b32)
#    define HAVE_ASYNC_LDS 1
#  endif
#endif
#ifndef HAVE_ASYNC_LDS
#  define HAVE_ASYNC_LDS 0
#endif

#if HAVE_ASYNC_LDS
typedef int __attribute__((address_space(1))) g1int;   // global-AS int (diag: '__device__ int *')
typedef int __attribute__((address_space(3))) l3int;   // LDS-AS int
__device__ __forceinline__ void async_copy_b32(const float* src, float* lds_dst) {
  __builtin_amdgcn_global_load_async_to_lds_b32(
      (g1int*)(uintptr_t)src,
      (l3int*)(unsigned)(uintptr_t)lds_dst, 0, 0);
}
__device__ __forceinline__ void wait_async() {
#  if __has_builtin(__builtin_amdgcn_s_wait_asynccnt)
  __builtin_amdgcn_s_wait_asynccnt(0);
#  else
  asm volatile("s_wait_asynccnt 0x0" ::: "memory");
#  endif
}
#endif

// ---------------------------------------------------------------- K0: spatial sum of source
__global__ void __launch_bounds__(256)
rowsum_kernel(const float* __restrict__ src, float* __restrict__ ssum) {
  __shared__ float red[256];
  const int row = blockIdx.x;                    // b*256 + c
  const float* p = src + (size_t)row * 4096;
  float s = 0.f;
  for (int i = threadIdx.x; i < 4096; i += 256) s += p[i];
  red[threadIdx.x] = s;
  __syncthreads();
  for (int w = 128; w > 0; w >>= 1) {
    if (threadIdx.x < w) red[threadIdx.x] += red[threadIdx.x + w];
    __syncthreads();
  }
  if (threadIdx.x == 0) ssum[row] = red[0];
}

// ---------------------------------------------------------------- K1: fold weights -> G, e
__global__ void __launch_bounds__(256)
setup_kernel(const float* __restrict__ key_w,  const float* __restrict__ key_b,
             const float* __restrict__ query_w, const float* __restrict__ query_b,
             const float* __restrict__ lin_w,   const float* __restrict__ lin_b,
             const float* __restrict__ attn_w,  const float* __restrict__ attn_b,
             const float* __restrict__ gamma_p, const float* __restrict__ ssum,
             float* __restrict__ G, float* __restrict__ e)
{
  __shared__ float sMk[32 * 256];   // Mk = Wk @ key_w
  __shared__ float sqraw[4 * 32];
  __shared__ float sqs[4 * 32];
  __shared__ float skb2[32];        // Wk @ key_b
  const int t = threadIdx.x;
  const float gamma = gamma_p[0];

  { // Mk[a][c]
    const int a  = t >> 3;
    const int c0 = (t & 7) * 32;
    for (int c = c0; c < c0 + 32; ++c) {
      float s = 0.f;
      for (int j = 0; j < 32; ++j) s += lin_w[a * 64 + j] * key_w[j * 256 + c];
      sMk[a * 256 + c] = s;
    }
  }
  if (t < 32) { // kb2
    float s = 0.f;
    for (int j = 0; j < 32; ++j) s += lin_w[t * 64 + j] * key_b[j];
    skb2[t] = s;
  }
  if (t < 128) { // qraw[b][a] = query_w @ ssum[b] + WH*query_b
    const int b = t >> 5, a = t & 31;
    float s = 0.f;
    for (int c = 0; c < 256; ++c) s += query_w[a * 256 + c] * ssum[b * 256 + c];
    sqraw[b * 32 + a] = s + WH_F * query_b[a];
  }
  __syncthreads();
  if (t < 128) { // qs[b][o] = Wq @ qraw[b]
    const int b = t >> 5, o = t & 31;
    float s = 0.f;
    for (int a = 0; a < 32; ++a) s += lin_w[o * 64 + 32 + a] * sqraw[b * 32 + a];
    sqs[b * 32 + o] = s;
  }
  __syncthreads();
  { // G row o = t, and e[b][o]
    const int o = t;
    for (int c = 0; c < 256; ++c) {
      float s = 0.f;
      for (int a = 0; a < 32; ++a) s += attn_w[o * 288 + 256 + a] * sMk[a * 256 + c];
      const float aeff = attn_w[o * 288 + c] + WH_F * s;
      G[o * 256 + c] = ((o == c) ? 1.0f : 0.0f) + gamma * aeff;
    }
    for (int b = 0; b < 4; ++b) {
      float s = 0.f;
      for (int a = 0; a < 32; ++a)
        s += attn_w[o * 288 + 256 + a] * (WH_F * (lin_b[a] + skb2[a]) + sqs[b * 32 + a]);
      e[b * 256 + o] = gamma * (s + attn_b[o]);
    }
  }
}

// ---------------------------------------------------------------- K1b: repack conv weights
// Arep[(kh*3+kw)][o][ci] <- conv_w[o][ci][kh][kw]  (A-fragment: contiguous K pair per lane)
__global__ void __launch_bounds__(256)
repack_w_kernel(const float* __restrict__ cw, float* __restrict__ Arep) {
  const int idx = blockIdx.x * 256 + threadIdx.x;       // dest index
  if (idx >= 128 * 512 * 9) return;
  const int ci   = idx & 511;
  const int rest = idx >> 9;        // = k9*128 + o
  const int o    = rest & 127;
  const int k9   = rest >> 7;
  Arep[idx] = cw[((size_t)o * 512 + ci) * 9 + k9];
}

// ---------------------------------------------------------------- K2: x1 = G @ target + e  (WMMA f32)
__global__ void __launch_bounds__(256)
gemm_x1_kernel(const float* __restrict__ G, const float* __restrict__ e,
               const float* __restrict__ target, float* __restrict__ x1)
{
  const int lane = threadIdx.x & 31;
  const int wave = threadIdx.x >> 5;
  const int half = lane >> 4;       // 0: lanes 0-15, 1: lanes 16-31
  const int l16  = lane & 15;
  const int mBase = blockIdx.y * 16;
  const int nBase = blockIdx.x * 128 + wave * 16;
  const int b = blockIdx.z;
  const float* T = target + (size_t)b * 256 * 4096;

  const int m = mBase + l16;
  const int n = nBase + l16;
  v8f acc = {};
  for (int k = 0; k < 256; k += 4) {
    const int ka = k + 2 * half;
    v2f A = *(const v2f*)(G + (size_t)m * 256 + ka);      // A(16x4): contiguous K pair
    v2f B;                                                // B(4x16): rows ka, ka+1
    B.x = T[(size_t)ka * 4096 + n];
    B.y = T[(size_t)(ka + 1) * 4096 + n];
    acc = __builtin_amdgcn_wmma_f32_16x16x4_f32(false, A, false, B, (short)0, acc,
                                                false, false);
  }
  for (int r = 0; r < 8; ++r) {
    const int row = mBase + r + 8 * half;
    x1[((size_t)b * 256 + row) * 4096 + n] = acc[r] + e[b * 256 + row];
  }
}

// ---------------------------------------------------------------- K3: implicit-GEMM 3x3 conv (WMMA f32)
// Block: one 16-pixel row tile x all 128 out channels (8 waves x 16 oc).
// Activations (18 cols x 3 rows x 512 ch, zero-padded) staged in LDS in two
// 256-channel stages; stage 1 reads prev with nearest-upsample indexing.
#define TILE_ELEMS (256 * 54)   // 54 = 3 rows * 18 cols per channel
__global__ void __launch_bounds__(256)
conv_kernel(const float* __restrict__ x1, const float* __restrict__ prev,
            const float* __restrict__ Arep, float* __restrict__ y)
{
  __shared__ float tile[TILE_ELEMS];              // 55296 bytes
  const int t = threadIdx.x;
  const int lane = t & 31;
  const int wave = t >> 5;                        // o-tile 0..7
  const int half = lane >> 4;
  const int l16  = lane & 15;
  const int b  = blockIdx.z;
  const int hh = blockIdx.x >> 2;                 // output row 0..63
  const int wBase = (blockIdx.x & 3) << 4;        // 16-pixel column tile
  const int oBase = wave * 16;

  v8f acc = {};
  for (int stage = 0; stage < 2; ++stage) {
    __syncthreads();   // previous stage's LDS reads are done before overwrite
    // ---- stage activations: tile[ci*54 + r*18 + c], zero-padded halo
    for (int i = t; i < TILE_ELEMS; i += 256) {
      const int ci  = i / 54;
      const int rem = i - ci * 54;
      const int r   = rem / 18;
      const int c   = rem - r * 18;
      const int row = hh + r - 1;
      const int col = wBase + c - 1;
      const bool ok = (row >= 0) && (row < 64) && (col >= 0) && (col < 64);
      const float* src = (stage == 0)
          ? x1  + ((size_t)b * 256 + ci) * 4096 + row * 64 + col
          : prev + ((size_t)b * 256 + ci) * 1024 + (row >> 1) * 32 + (col >> 1);
#if HAVE_ASYNC_LDS
      if (ok) async_copy_b32(src, &tile[i]);
      else    tile[i] = 0.f;
#else
      tile[i] = ok ? *src : 0.f;
#endif
    }
#if HAVE_ASYNC_LDS
    wait_async();
#endif
    __syncthreads();
    // ---- compute: per wave, 16 oc x 16 px, K = 256 channels of this stage x 9 taps
    for (int kh = 0; kh < 3; ++kh) {
      for (int kw = 0; kw < 3; ++kw) {
        const float* wp = Arep + ((size_t)((kh * 3 + kw) * 128) + oBase + l16) * 512
                               + stage * 256;
        const float* bp = &tile[kh * 18 + l16 + kw];
        for (int c = 0; c < 256; c += 4) {
          const int ci = c + 2 * half;
          v2f A = *(const v2f*)(wp + ci);          // contiguous b64
          v2f B;
          B.x = bp[ci * 54];                        // ds_load_b32
          B.y = bp[(ci + 1) * 54];
          acc = __builtin_amdgcn_wmma_f32_16x16x4_f32(false, A, false, B, (short)0,
                                                      acc, false, false);
        }
      }
    }
  }
  const int n = hh * 64 + wBase + l16;
  for (int r = 0; r < 8; ++r) {
    const int orow = oBase + r + 8 * half;
    y[((size_t)b * 128 + orow) * 4096 + n] = acc[r];  // conv_b omitted: cancels in norm
  }
}

// ---------------------------------------------------------------- K4: instance norm + ReLU
__global__ void __launch_bounds__(256)
norm_relu_kernel(const float* __restrict__ y, float* __restrict__ out)
{
  __shared__ float s1[256], s2[256];
  const int row = blockIdx.x;                    // b*128 + o
  const float* p = y + (size_t)row * 4096;
  float a = 0.f, sq = 0.f;
  for (int i = threadIdx.x; i < 4096; i += 256) { const float v = p[i]; a += v; sq += v * v; }
  s1[threadIdx.x] = a; s2[threadIdx.x] = sq;
  __syncthreads();
  for (int w = 128; w > 0; w >>= 1) {
    if (threadIdx.x < w) {
      s1[threadIdx.x] += s1[threadIdx.x + w];
      s2[threadIdx.x] += s2[threadIdx.x + w];
    }
    __syncthreads();
  }
  const float mean  = s1[0] * (1.0f / 4096.0f);
  const float var   = s2[0] * (1.0f / 4096.0f) - mean * mean;
  const float scale = rsqrtf(var + 1e-5f);
  float* q = out + (size_t)row * 4096;
  for (int i = threadIdx.x; i < 4096; i += 256) {
    const float v = (p[i] - mean) * scale;
    q[i] = v > 0.f ? v : 0.f;
  }
}

// ---------------------------------------------------------------- launch
extern "C" void kernel_launch(void* const* d_in, const int* in_sizes, int n_in,
                              void* d_out, int out_size, void* d_ws, size_t ws_size,
                              hipStream_t stream) {
  const float* source  = (const float*)d_in[0];
  const float* target  = (const float*)d_in[1];
  const float* prev    = (const float*)d_in[2];
  const float* key_w   = (const float*)d_in[3];
  const float* key_b   = (const float*)d_in[4];
  const float* query_w = (const float*)d_in[5];
  const float* query_b = (const float*)d_in[6];
  const float* lin_w   = (const float*)d_in[7];
  const float* lin_b   = (const float*)d_in[8];
  const float* attn_w  = (const float*)d_in[9];
  const float* attn_b  = (const float*)d_in[10];
  const float* gamma   = (const float*)d_in[11];
  const float* conv_w  = (const float*)d_in[12];
  // d_in[13] conv_b: unused — exactly cancelled by instance-norm mean subtraction.
  (void)in_sizes; (void)n_in; (void)out_size; (void)ws_size;

  float* G    = (float*)d_ws;
  float* e    = G + 65536;
  float* ssum = e + 1024;
  float* Arep = ssum + 1024;
  float* x1   = Arep + 9 * 128 * 512;
  float* y    = x1 + 4 * 256 * 4096;

  rowsum_kernel<<<dim3(1024), dim3(256), 0, stream>>>(source, ssum);
  setup_kernel<<<dim3(1), dim3(256), 0, stream>>>(key_w, key_b, query_w, query_b,
                                                  lin_w, lin_b, attn_w, attn_b,
                                                  gamma, ssum, G, e);
  repack_w_kernel<<<dim3((128 * 512 * 9 + 255) / 256), dim3(256), 0, stream>>>(conv_w, Arep);
  gemm_x1_kernel<<<dim3(32, 16, 4), dim3(256), 0, stream>>>(G, e, target, x1);
  conv_kernel<<<dim3(256, 1, 4), dim3(256), 0, stream>>>(x1, prev, Arep, y);
  norm_relu_kernel<<<dim3(512), dim3(256), 0, stream>>>(y, (float*)d_out);
}